// image_motif_57372173140487
// MI455X (gfx1250) — compile-verified
//
#include <hip/hip_runtime.h>
#include <math.h>

typedef __attribute__((ext_vector_type(2))) float v2f;
typedef __attribute__((ext_vector_type(8))) float v8f;

#define B_    2
#define CT    256
#define Hh    192
#define Ww    192
#define Gn    16
#define Cg    16
#define Pn    144            // Cg*9
#define Ho_   190
#define Wo_   190
#define Ln    (190*190)      // 36100
#define LC    64             // L-chunk staged in LDS
#define STR   66             // padded row stride (floats) -> conflict-free ds_load_b64
#define NCH   ((Ln + LC - 1)/LC)     // 565 chunks total
#define NSEG  8                      // K-reduction split across blocks
#define CPS   ((NCH + NSEG - 1)/NSEG) // 71 chunks per segment
#define GRAM_ELEMS (Pn*Pn)           // 20736
#define NBG   32                     // (b,g) pairs
#define PART_TOTAL (NSEG*NBG*GRAM_ELEMS) // partial grams (floats)
#define GRAM_TOTAL (NBG*GRAM_ELEMS)      // final grams (floats)
// ws layout: [0, PART_TOTAL) partials | [PART_TOTAL, +GRAM_TOTAL) final | params

// ---------------------------------------------------------------------------
// Phase 1: partial gram[seg,b,g] = motif(Lseg) . motif(Lseg)^T via fp32 WMMA.
// Grid = 32 (b,g) x 8 K-segments = 256 blocks; 9 waves; wave w owns tile row.
// ---------------------------------------------------------------------------
__global__ __launch_bounds__(288)
void motif_gram_kernel(const float* __restrict__ feat, float* __restrict__ ws) {
    __shared__ __align__(16) float tile[Pn * STR];

    const int blk  = blockIdx.x;        // 0..255
    const int bg   = blk & (NBG - 1);   // 0..31
    const int seg  = blk >> 5;          // 0..7
    const int b    = bg >> 4;
    const int g    = bg & 15;
    const int tid  = threadIdx.x;
    const int wave = tid >> 5;          // 0..8  (tile row tm)
    const int lane = tid & 31;
    const int lane16 = lane & 15;
    const int half   = lane >> 4;       // K sub-block select per ISA layout

    v8f zero = {};
    v8f acc[9];
#pragma unroll
    for (int t = 0; t < 9; ++t) acc[t] = zero;

    const float* fbase = feat + (size_t)(b * CT + g * Cg) * (Hh * Ww);

    const int ch_beg = seg * CPS;
    const int ch_end = min(ch_beg + CPS, NCH);

    for (int ch = ch_beg; ch < ch_end; ++ch) {
        const int l0 = ch * LC;
        // ---- stage 144 x 64 motif chunk into LDS (9216 elems / 288 thr = 32 each)
        for (int idx = tid; idx < Pn * LC; idx += 288) {
            int p   = idx >> 6;          // / LC
            int col = idx & (LC - 1);
            int l   = l0 + col;
            float v = 0.0f;
            if (l < Ln) {
                int c   = p / 9;
                int rem = p - c * 9;
                int i   = rem / 3;
                int j   = rem - i * 3;
                int ho  = l / Wo_;
                int wo  = l - ho * Wo_;
                v = fbase[(size_t)c * (Hh * Ww) + (ho + i) * Ww + (wo + j)];
            }
            tile[p * STR + col] = v;
        }
        __syncthreads();

        // ---- 16 k-steps of V_WMMA_F32_16X16X4_F32 per tile
        const float* arow = &tile[(wave * 16 + lane16) * STR];
#pragma unroll 4
        for (int kk = 0; kk < LC / 4; ++kk) {
            const int cbase = kk * 4 + half * 2;   // A: VGPR0=K0/K2, VGPR1=K1/K3
            v2f afrag = *(const v2f*)(arow + cbase);
#pragma unroll
            for (int tn = 0; tn < 9; ++tn) {
                v2f bfrag = *(const v2f*)(&tile[(tn * 16 + lane16) * STR + cbase]);
                acc[tn] = __builtin_amdgcn_wmma_f32_16x16x4_f32(
                    false, afrag, false, bfrag, (short)0, acc[tn], false, false);
            }
        }
        __syncthreads();
    }

    // ---- write partial gram: C/D layout: VGPR r -> M = r + half*8, N = lane16
    float* gout = ws + ((size_t)seg * NBG + bg) * GRAM_ELEMS;
#pragma unroll
    for (int tn = 0; tn < 9; ++tn) {
#pragma unroll
        for (int r = 0; r < 8; ++r) {
            int p = wave * 16 + r + half * 8;
            int q = tn * 16 + lane16;
            gout[p * Pn + q] = acc[tn][r];
        }
    }
}

// ---------------------------------------------------------------------------
// Phase 1b: deterministic fixed-order sum of the 8 partial grams.
// ---------------------------------------------------------------------------
__global__ __launch_bounds__(256)
void gram_sum_kernel(float* __restrict__ ws) {
    const size_t idx = (size_t)blockIdx.x * 256 + threadIdx.x;
    if (idx >= (size_t)NBG * GRAM_ELEMS) return;
    float s = 0.0f;
#pragma unroll
    for (int seg = 0; seg < NSEG; ++seg)
        s += ws[(size_t)seg * NBG * GRAM_ELEMS + idx];
    ws[(size_t)PART_TOTAL + idx] = s;
}

// ---------------------------------------------------------------------------
// Phase 2: per group: argmin distances (clamped d2; sqrt monotone -> skipped),
// bincount over both batches, top-3 (strict-greater scan = lowest-index ties).
// ---------------------------------------------------------------------------
__global__ __launch_bounds__(256)
void motif_select_kernel(float* __restrict__ ws) {
    __shared__ float diag[2][Pn];
    __shared__ int   mini[2][Pn];
    __shared__ int   counts[Pn];

    const int g   = blockIdx.x;
    const int tid = threadIdx.x;
    float* gram = ws + PART_TOTAL;

    for (int t = tid; t < 2 * Pn; t += 256) {
        int b = t / Pn, p = t - b * Pn;
        diag[b][p] = gram[(size_t)(b * 16 + g) * GRAM_ELEMS + p * Pn + p];
    }
    if (tid < Pn) counts[tid] = 0;
    __syncthreads();

    for (int t = tid; t < 2 * Pn; t += 256) {
        int b = t / Pn, p = t - b * Pn;
        const float* row = gram + (size_t)(b * 16 + g) * GRAM_ELEMS + p * Pn;
        float dp = diag[b][p];
        float best = INFINITY;
        int bq = 0;
        for (int q = 0; q < Pn; ++q) {
            if (q == p) continue;
            float d2 = fmaxf(dp + diag[b][q] - 2.0f * row[q], 0.0f);
            if (d2 < best) { best = d2; bq = q; }   // first-index tie-break
        }
        mini[b][p] = bq;
    }
    __syncthreads();

    for (int t = tid; t < 2 * Pn; t += 256) {
        int b = t / Pn, p = t - b * Pn;
        atomicAdd(&counts[mini[b][p]], 1);
    }
    __syncthreads();

    if (tid == 0) {
        int chosen[3]; float cf[3]; float total = 0.0f;
        for (int o = 0; o < 3; ++o) {
            int bc = -1, bi = 0;
            for (int p = 0; p < Pn; ++p) {
                bool used = false;
                for (int u = 0; u < o; ++u) used |= (chosen[u] == p);
                if (!used && counts[p] > bc) { bc = counts[p]; bi = p; }
            }
            chosen[o] = bi; cf[o] = (float)bc; total += (float)bc;
        }
        float* prm  = ws + PART_TOTAL + GRAM_TOTAL + g * 8;
        int*   prmi = (int*)prm;
        prmi[0] = chosen[0]; prmi[1] = chosen[1]; prmi[2] = chosen[2];
        prm[3] = cf[0]; prm[4] = cf[1]; prm[5] = cf[2]; prm[6] = total;
    }
}

// ---------------------------------------------------------------------------
// Phase 3: fused motif_new + fold3, collapsed to a streaming pass:
// out[b,c,h,w] = sum_{i,j valid} sum_o floor( x * r_o[h-i,w-j] * cf_o / total )
// ---------------------------------------------------------------------------
__global__ __launch_bounds__(256)
void motif_fold_kernel(const float* __restrict__ feat,
                       const float* __restrict__ ws,
                       float* __restrict__ out) {
    const int idx = blockIdx.x * 256 + threadIdx.x;
    int w  = idx % Ww;
    int t  = idx / Ww;
    int h  = t % Hh;  t /= Hh;
    int ch = t % CT;
    int b  = t / CT;
    int g  = ch >> 4;

    const float* prm  = ws + PART_TOTAL + GRAM_TOTAL + g * 8;
    const int*   prmi = (const int*)prm;
    float cfv[3] = { prm[3], prm[4], prm[5] };
    float total  = prm[6];

    const float* rp[3];
#pragma unroll
    for (int o = 0; o < 3; ++o) {
        int ti  = prmi[o];
        int c   = ti / 9;
        int rem = ti - c * 9;
        int io  = rem / 3;
        int jo  = rem - io * 3;
        rp[o] = feat + (size_t)((b * CT + g * Cg + c) * Hh + io) * Ww + jo;
    }

    float x = feat[idx];
    float acc = 0.0f;
#pragma unroll
    for (int di = 0; di < 3; ++di) {
        int ho = h - di;
        if ((unsigned)ho >= Ho_) continue;
#pragma unroll
        for (int dj = 0; dj < 3; ++dj) {
            int wo = w - dj;
            if ((unsigned)wo >= Wo_) continue;
            int off = ho * Ww + wo;
#pragma unroll
            for (int o = 0; o < 3; ++o) {
                float r = rp[o][off];
                acc += floorf(((x * r) * cfv[o]) / total);  // reference op order
            }
        }
    }
    out[idx] = acc;
}

extern "C" void kernel_launch(void* const* d_in, const int* in_sizes, int n_in,
                              void* d_out, int out_size, void* d_ws, size_t ws_size,
                              hipStream_t stream) {
    const float* feat = (const float*)d_in[0];
    float* out = (float*)d_out;
    float* ws  = (float*)d_ws;  // needs (8+1)*32*144*144 + 16*8 floats (~23.9 MB)

    hipLaunchKernelGGL(motif_gram_kernel, dim3(NSEG * NBG), dim3(288), 0, stream,
                       feat, ws);
    hipLaunchKernelGGL(gram_sum_kernel,
                       dim3((NBG * GRAM_ELEMS + 255) / 256), dim3(256), 0, stream, ws);
    hipLaunchKernelGGL(motif_select_kernel, dim3(16), dim3(256), 0, stream, ws);
    hipLaunchKernelGGL(motif_fold_kernel, dim3((out_size + 255) / 256), dim3(256),
                       0, stream, feat, ws, out);
}